// GHMC_Loss_36155034697956
// MI455X (gfx1250) — compile-verified
//
#include <hip/hip_runtime.h>
#include <stdint.h>

// ---- problem constants (fixed by the harness setup_inputs) ----
#define BATCH       8
#define NCLASS      19
#define H_          512
#define W_          1024
#define HW          (H_ * W_)        // 524288 = 2^19
#define HW_LOG2     19
#define NPIX        (BATCH * HW)     // 4194304
#define NBIN        (HW / 10)        // 52428  (fits in uint16)
#define TPB         256
#define PIX_PER_WAVE 128             // 32 lanes x b128 (4 floats/lane)
#define PIX_PER_BLK  (PIX_PER_WAVE * (TPB / 32))   // 1024
#define NBLK_PIX     (NPIX / PIX_PER_BLK)          // 4096
#define NBLK_RED     (NPIX / TPB)                  // 16384

// LDS byte-offset of a generic pointer into __shared__ memory: generic LDS
// addresses carry the workgroup-relative offset in the low 32 bits, which is
// exactly what the async VDST VGPR expects (an offset added to the LDS base).
__device__ __forceinline__ uint32_t lds_off_u32(const void* p) {
    return (uint32_t)(uintptr_t)p;
}

// ---------------------------------------------------------------------------
// Kernel 1: zero histogram (graph replays -> must re-zero every call)
// ---------------------------------------------------------------------------
__global__ __launch_bounds__(TPB) void ghmc_zero(uint32_t* __restrict__ hist,
                                                 int n) {
    int i = blockIdx.x * TPB + threadIdx.x;
    if (i < n) hist[i] = 0u;
}

// ---------------------------------------------------------------------------
// Kernel 2: per-pixel log-softmax + bin + histogram.
// Each wave async-stages 128 consecutive pixels x 19 channels into LDS via
// global_load_async_to_lds_b128 (512B per wave-instruction, NT so the
// streamed 318MB of x doesn't evict the reused hist/nll/bins in L2), then
// each thread computes 4 softmaxes from LDS using hardware v_exp/v_log.
// ---------------------------------------------------------------------------
__global__ __launch_bounds__(TPB) void ghmc_pixel_pass(
    const float* __restrict__ x, const int* __restrict__ target,
    uint32_t* __restrict__ hist, float* __restrict__ nll_out,
    uint16_t* __restrict__ bin_out) {
    // [wave][channel][pixel-in-wave] : 8 * 19 * 128 * 4B = 77824 B
    __shared__ float smem[TPB / 32][NCLASS][PIX_PER_WAVE];

    const int tid     = threadIdx.x;
    const int wave    = tid >> 5;
    const int lane    = tid & 31;
    const int wavePix = blockIdx.x * PIX_PER_BLK + wave * PIX_PER_WAVE;
    const int b       = wavePix >> HW_LOG2;      // block never straddles b
    const int hw      = wavePix & (HW - 1);

    // ---- async stage: 19 x b128 fire-and-forget copies per wave ----
    const float* gp = x + (size_t)b * NCLASS * HW + hw;
    uint64_t ga = (uint64_t)(uintptr_t)(gp + lane * 4);  // 16B per lane
    uint32_t la = lds_off_u32(&smem[wave][0][lane * 4]);
#pragma unroll
    for (int c = 0; c < NCLASS; ++c) {
        asm volatile("global_load_async_to_lds_b128 %0, %1, off th:TH_LOAD_NT"
                     :
                     : "v"(la), "v"(ga)
                     : "memory");
        ga += (uint64_t)HW * 4u;           // next channel plane (2 MB stride)
        la += PIX_PER_WAVE * 4u;           // next LDS channel row
    }
    // drain ASYNCcnt: the wave's full 19x128 tile is now in LDS
    asm volatile("s_wait_asynccnt 0x0" ::: "memory");

    // ---- 4 pixels per thread, pixel-major in LDS (conflict-free) ----
#pragma unroll
    for (int j = 0; j < 4; ++j) {
        const int pinw = lane + 32 * j;          // pixel index within wave tile
        const int pix  = wavePix + pinw;
        const int t    = target[pix];

        float v[NCLASS];
        float m = -3.402823466e38f;
#pragma unroll
        for (int c = 0; c < NCLASS; ++c) {
            v[c] = smem[wave][c][pinw];
            m = fmaxf(m, v[c]);
        }
        // native v_exp_f32: ~2 VALU ops per exp, keeps the pass memory-bound
        float s = 0.0f, vt = 0.0f;
#pragma unroll
        for (int c = 0; c < NCLASS; ++c) {
            s += __expf(v[c] - m);
            vt = (c == t) ? v[c] : vt;  // select, never dynamic-index
        }
        const float logp_t = vt - m - __logf(s);   // native v_log_f32
        const float nll    = -logp_t;
        const float pt     = __expf(logp_t);
        const float g      = fabsf(pt - 1.0f);

        int bin = (int)floorf(g * ((float)NBIN - 1e-4f));
        bin = bin < 0 ? 0 : (bin > NBIN - 1 ? NBIN - 1 : bin);

        atomicAdd(&hist[b * NBIN + bin], 1u);  // non-returning L2 atomic
        nll_out[pix] = nll;
        bin_out[pix] = (uint16_t)bin;
    }
}

// ---------------------------------------------------------------------------
// Kernel 3: per-batch count of non-empty bins (8 blocks, one per batch elem)
// ---------------------------------------------------------------------------
__global__ __launch_bounds__(TPB) void ghmc_nonempty(
    const uint32_t* __restrict__ hist, float* __restrict__ nonempty) {
    __shared__ uint32_t red[TPB];
    const int b = blockIdx.x;
    uint32_t cnt = 0;
    for (int i = threadIdx.x; i < NBIN; i += TPB)
        cnt += (hist[b * NBIN + i] != 0u) ? 1u : 0u;
    red[threadIdx.x] = cnt;
    __syncthreads();
    for (int off = TPB / 2; off > 0; off >>= 1) {
        if (threadIdx.x < off) red[threadIdx.x] += red[threadIdx.x + off];
        __syncthreads();
    }
    if (threadIdx.x == 0) nonempty[b] = (float)red[0];
}

// ---------------------------------------------------------------------------
// Kernel 4: weight gather + per-block tree reduction -> partial sums
// weight = N / max(count * nonempty, 1); hist/nll/bins should be L2-resident
// ---------------------------------------------------------------------------
__global__ __launch_bounds__(TPB) void ghmc_weight_sum(
    const uint32_t* __restrict__ hist, const float* __restrict__ nonempty,
    const float* __restrict__ nll, const uint16_t* __restrict__ bins,
    float* __restrict__ partials) {
    __shared__ float red[TPB];
    const int pix = blockIdx.x * TPB + threadIdx.x;
    const int b   = pix >> HW_LOG2;
    const float ne = nonempty[b];
    const uint32_t cnt = hist[b * NBIN + (int)bins[pix]];
    const float gd = fmaxf((float)cnt * ne, 1.0f);
    const float w  = (float)HW / gd;  // N = H*W
    red[threadIdx.x] = nll[pix] * w;
    __syncthreads();
    for (int off = TPB / 2; off > 0; off >>= 1) {
        if (threadIdx.x < off) red[threadIdx.x] += red[threadIdx.x + off];
        __syncthreads();
    }
    if (threadIdx.x == 0) partials[blockIdx.x] = red[0];
}

// ---------------------------------------------------------------------------
// Kernel 5: single-block deterministic final reduction -> mean
// ---------------------------------------------------------------------------
__global__ __launch_bounds__(1024) void ghmc_final(
    const float* __restrict__ partials, float* __restrict__ out, int n) {
    __shared__ float red[1024];
    float s = 0.0f;
    for (int i = threadIdx.x; i < n; i += 1024) s += partials[i];
    red[threadIdx.x] = s;
    __syncthreads();
    for (int off = 512; off > 0; off >>= 1) {
        if (threadIdx.x < off) red[threadIdx.x] += red[threadIdx.x + off];
        __syncthreads();
    }
    if (threadIdx.x == 0) out[0] = red[0] / (float)NPIX;
}

// ---------------------------------------------------------------------------
extern "C" void kernel_launch(void* const* d_in, const int* in_sizes, int n_in,
                              void* d_out, int out_size, void* d_ws,
                              size_t ws_size, hipStream_t stream) {
    (void)in_sizes; (void)n_in; (void)out_size; (void)ws_size;
    const float* x      = (const float*)d_in[0];
    const int*   target = (const int*)d_in[1];  // harness: integer -> int32
    float* out = (float*)d_out;

    // workspace layout (256B aligned slices), ~27 MB total
    char* ws = (char*)d_ws;
    size_t off = 0;
    uint32_t* hist = (uint32_t*)(ws + off);
    off += ((size_t)BATCH * NBIN * sizeof(uint32_t) + 255) & ~(size_t)255;
    float* nonempty = (float*)(ws + off);
    off += 256;
    float* nll = (float*)(ws + off);
    off += ((size_t)NPIX * sizeof(float) + 255) & ~(size_t)255;
    uint16_t* bins = (uint16_t*)(ws + off);
    off += ((size_t)NPIX * sizeof(uint16_t) + 255) & ~(size_t)255;
    float* partials = (float*)(ws + off);
    off += ((size_t)NBLK_RED * sizeof(float) + 255) & ~(size_t)255;

    const int nhist = BATCH * NBIN;
    ghmc_zero<<<(nhist + TPB - 1) / TPB, TPB, 0, stream>>>(hist, nhist);
    ghmc_pixel_pass<<<NBLK_PIX, TPB, 0, stream>>>(x, target, hist, nll, bins);
    ghmc_nonempty<<<BATCH, TPB, 0, stream>>>(hist, nonempty);
    ghmc_weight_sum<<<NBLK_RED, TPB, 0, stream>>>(hist, nonempty, nll, bins,
                                                  partials);
    ghmc_final<<<1, 1024, 0, stream>>>(partials, out, NBLK_RED);
}